// pose3d_online_45707041964541
// MI455X (gfx1250) — compile-verified
//
#include <hip/hip_runtime.h>
#include <hip/hip_bf16.h>
#include <stdint.h>

// ---------------------------------------------------------------------------
// Pose loss, fused single pass.  W=131072 windows, J=21 joints, B=20 bones.
// Bandwidth-bound (~103 MB read -> ~4.4us floor at 23.3 TB/s).
// CDNA5 paths: 2x TDM tensor_load_to_lds (pose halo tile + bone2d tile) under
// one s_wait_tensorcnt, and a V_WMMA_F32_16X16X4_F32 wave-reduction.
// ---------------------------------------------------------------------------

#define WN 131072
#define JN 21
#define BN 20
#define BLK 128              // windows per block
#define ROWS (BLK + 2)       // pose rows staged per block (halo of 2)
#define ROWF 63              // floats per pose row (3*21)
#define B2F  42              // floats per bone2d row (2*21)
#define NBLK (WN / BLK)      // 1024 blocks

typedef unsigned int u32x4 __attribute__((ext_vector_type(4)));
typedef int          i32x4 __attribute__((ext_vector_type(4)));
typedef int          i32x8 __attribute__((ext_vector_type(8)));
typedef float        f32x2 __attribute__((ext_vector_type(2)));
typedef float        f32x8 __attribute__((ext_vector_type(8)));

#if __has_builtin(__builtin_amdgcn_tensor_load_to_lds)
#define HAVE_TDM 1
// 1-D contiguous TDM load: nelem_tile elements from gaddr -> LDS offset.
// Elements beyond nelem_tensor are zero-filled by TDM OOB handling.
__device__ __forceinline__ void tdm_load_1d(unsigned long long gaddr,
                                            unsigned lds_off,
                                            unsigned nelem_tile,
                                            unsigned nelem_tensor) {
    // D# group 0: count=1 | lds_addr | global_addr lo | global_addr hi + type=2
    u32x4 g0 = { 1u,
                 lds_off,
                 (unsigned)(gaddr & 0xFFFFFFFFull),
                 (unsigned)((gaddr >> 32) & 0x1FFFFFFull) | 0x80000000u };
    // D# group 1: data_size=4B; tensor_dim0=nelem_tensor; tensor_dim1=1;
    //             tile_dim0=nelem_tile; tile_dim1=1; tensor_dim0_stride=tile.
    i32x8 g1 = { (int)0x00020000u,
                 (int)((nelem_tensor & 0xFFFFu) << 16),
                 (int)((nelem_tensor >> 16) | (1u << 16)),
                 (int)(nelem_tile << 16),
                 1,
                 (int)nelem_tile,
                 0, 0 };
    i32x4 gz = { 0, 0, 0, 0 };
#if defined(__clang_major__) && (__clang_major__ >= 23)
    i32x8 gz8 = { 0, 0, 0, 0, 0, 0, 0, 0 };
    __builtin_amdgcn_tensor_load_to_lds(g0, g1, gz, gz, gz8, 0);
#else
    __builtin_amdgcn_tensor_load_to_lds(g0, g1, gz, gz, 0);
#endif
}
#endif

__global__ __launch_bounds__(BLK)
void pose_loss_stage1(const float* __restrict__ pose,     // (W+1,3,21)
                      const float* __restrict__ bone2d,   // (W,2,21)
                      const float* __restrict__ lift,     // (W,3,20)
                      const float* __restrict__ Rd,       // (W,3,3)
                      const float* __restrict__ Cd,       // (W,3,1)
                      const float* __restrict__ blen,     // (20)
                      const float* __restrict__ Rcam,     // (3,3)
                      const int*   __restrict__ conn,     // (20,2)
                      float* __restrict__ partial)        // (NBLK)
{
    __shared__ float sP[ROWS * ROWF];    // 130*63*4 = 32760 bytes
    __shared__ float sB2[BLK * B2F];     // 128*42*4 = 21504 bytes
    __shared__ float sWave[BLK / 32];

    const int t  = threadIdx.x;
    const int w0 = blockIdx.x * BLK;
    const int w  = w0 + t;

    // ---- stage pose3d rows [w0, w0+129] and bone2d rows [w0, w0+127] ----
    const unsigned poseTile   = (unsigned)(ROWS * ROWF);           // 8190 elems
    const unsigned poseTotal  = (unsigned)(WN + 1) * (unsigned)ROWF;
    const unsigned poseBase   = (unsigned)w0 * (unsigned)ROWF;
    const unsigned poseRemain = poseTotal - poseBase;              // OOB -> 0
    const unsigned b2Tile     = (unsigned)(BLK * B2F);             // 5376 elems
    const unsigned b2Base     = (unsigned)w0 * (unsigned)B2F;
    const unsigned b2Remain   = (unsigned)WN * (unsigned)B2F - b2Base;

#ifdef HAVE_TDM
    if (t < 32) {
        tdm_load_1d((unsigned long long)(uintptr_t)pose + (unsigned long long)poseBase * 4ull,
                    (unsigned)(uintptr_t)(&sP[0]), poseTile, poseRemain);
        tdm_load_1d((unsigned long long)(uintptr_t)bone2d + (unsigned long long)b2Base * 4ull,
                    (unsigned)(uintptr_t)(&sB2[0]), b2Tile, b2Remain);
#if __has_builtin(__builtin_amdgcn_s_wait_tensorcnt)
        __builtin_amdgcn_s_wait_tensorcnt(0);
#endif
    }
#else
    // Fallback staging (also taken on the host compilation pass).
    for (unsigned i = (unsigned)t; i < poseTile; i += (unsigned)BLK)
        sP[i] = (i < poseRemain) ? pose[poseBase + i] : 0.0f;
    for (unsigned i = (unsigned)t; i < b2Tile; i += (unsigned)BLK)
        sB2[i] = bone2d[b2Base + i];
#endif
    __syncthreads();

    // ---- uniform (scalar) loads ----
    float rc[9];
#pragma unroll
    for (int i = 0; i < 9; ++i) rc[i] = Rcam[i];
    int   ib0[BN], ib1[BN];
    float bl[BN];
#pragma unroll
    for (int k = 0; k < BN; ++k) {
        ib0[k] = conn[2 * k + 0];
        ib1[k] = conn[2 * k + 1];
        bl[k]  = blen[k];
    }

    // ---- per-window data ----
    float rd[9];
    const float* rdp = Rd + (size_t)w * 9;
#pragma unroll
    for (int i = 0; i < 9; ++i) rd[i] = rdp[i];

    // M = R_cam @ R_drone^T  (einsum 'ij,wkj,...': contract j, so Rd transposed)
    float M[9];
#pragma unroll
    for (int i = 0; i < 3; ++i)
#pragma unroll
        for (int k = 0; k < 3; ++k)
            M[i * 3 + k] = rc[i * 3 + 0] * rd[k * 3 + 0] +
                           rc[i * 3 + 1] * rd[k * 3 + 1] +
                           rc[i * 3 + 2] * rd[k * 3 + 2];

    const float c0 = Cd[(size_t)w * 3 + 0];
    const float c1 = Cd[(size_t)w * 3 + 1];
    const float c2 = Cd[(size_t)w * 3 + 2];

    const float* p  = &sP[(t + 1) * ROWF];            // pose3d[w+1]
    const float* b2 = &sB2[t * B2F];                  // bone_2d[w]
    const float* lf = lift + (size_t)w * (3 * BN);

    // ---- projection loss ----
    float proj_acc = 0.0f;
#pragma unroll
    for (int j = 0; j < JN; ++j) {
        float x = p[j]          - c0;
        float y = p[JN + j]     - c1;
        float z = p[2 * JN + j] - c2;
        float cx = M[0] * x + M[1] * y + M[2] * z;
        float cy = M[3] * x + M[4] * y + M[5] * z;
        float cz = M[6] * x + M[7] * y + M[8] * z;
        float s  = 512.0f / cz;                 // F / z
        float du = (cx * s + 512.0f) - b2[j];          // PX
        float dv = (cy * s + 288.0f) - b2[JN + j];     // PY
        proj_acc += du * du + dv * dv;
    }

    // ---- lift loss ----
    float lift_acc = 0.0f;
#pragma unroll
    for (int k = 0; k < BN; ++k) {
        int i0 = ib0[k], i1 = ib1[k];
        float bx = p[i0]          - p[i1];
        float by = p[JN + i0]     - p[JN + i1];
        float bz = p[2 * JN + i0] - p[2 * JN + i1];
        float inv = 1.0f / (sqrtf(bx * bx + by * by + bz * bz) + 1e-8f);
        float dx = lf[k]          - bx * inv;
        float dy = lf[BN + k]     - by * inv;
        float dz = lf[2 * BN + k] - bz * inv;
        lift_acc += dx * dx + dy * dy + dz * dz;
    }

    // ---- bone-length loss: one pose row per thread (+ row W in last block) ----
    float bone_acc = 0.0f;
    {
        const float* q = &sP[t * ROWF];                // pose3d[w0+t]
#pragma unroll
        for (int k = 0; k < BN; ++k) {
            int i0 = ib0[k], i1 = ib1[k];
            float bx = q[i0]          - q[i1];
            float by = q[JN + i0]     - q[JN + i1];
            float bz = q[2 * JN + i0] - q[2 * JN + i1];
            float d  = bl[k] - (bx * bx + by * by + bz * bz);
            bone_acc += d * d;
        }
    }
    if (blockIdx.x == (unsigned)(NBLK - 1) && t == 0) {
        const float* q = &sP[BLK * ROWF];              // global row 131072
#pragma unroll
        for (int k = 0; k < BN; ++k) {
            int i0 = ib0[k], i1 = ib1[k];
            float bx = q[i0]          - q[i1];
            float by = q[JN + i0]     - q[JN + i1];
            float bz = q[2 * JN + i0] - q[2 * JN + i1];
            float d  = bl[k] - (bx * bx + by * by + bz * bz);
            bone_acc += d * d;
        }
    }

    // ---- smoothness loss: second difference at i = w (valid for w <= W-2) ----
    float smooth_acc = 0.0f;
    if (w <= WN - 2) {
        const float* a = &sP[t * ROWF];
        const float* b = &sP[(t + 1) * ROWF];
        const float* c = &sP[(t + 2) * ROWF];
        for (int i = 0; i < ROWF; ++i) {
            float dd = c[i] - 2.0f * b[i] + a[i];
            smooth_acc += dd * dd;
        }
    }

    float total = 0.25f * (proj_acc   * (1.0f / (2.0f * JN)) +
                           lift_acc   * (1.0f / (3.0f * JN)) +
                           bone_acc   * (1.0f / (JN - 1))    +
                           smooth_acc * (1.0f / (3.0f * JN)));

    // ---- wave32 reduction on the matrix pipe ----
    // A: VGPR0 = per-lane partial, VGPR1 = 0  =>  A[m][0]=v[m], A[m][2]=v[m+16]
    // B: all-ones 4x16            =>  D[m][n] = v[m] + v[m+16]  (exact)
    // D layout: lane l<16 holds rows 0..7 of column l in its
    // 8 VGPRs, lane l+16 holds rows 8..15; 8 adds + one shfl_xor(16) = total.
#if defined(__AMDGCN__) && __has_builtin(__builtin_amdgcn_wmma_f32_16x16x4_f32)
    {
        f32x2 a; a.x = total; a.y = 0.0f;
        f32x2 b; b.x = 1.0f;  b.y = 1.0f;
        f32x8 cc = {};
        f32x8 d = __builtin_amdgcn_wmma_f32_16x16x4_f32(
            /*neg_a=*/false, a, /*neg_b=*/false, b,
            /*c_mod=*/(short)0, cc, /*reuse_a=*/false, /*reuse_b=*/false);
        float s = ((d[0] + d[1]) + (d[2] + d[3])) + ((d[4] + d[5]) + (d[6] + d[7]));
        s += __shfl_xor(s, 16, 32);
        total = s;
    }
#else
#pragma unroll
    for (int off = 16; off > 0; off >>= 1)
        total += __shfl_down(total, off, 32);
#endif
    if ((t & 31) == 0) sWave[t >> 5] = total;
    __syncthreads();
    if (t == 0) {
        float s = 0.0f;
#pragma unroll
        for (int i = 0; i < BLK / 32; ++i) s += sWave[i];
        partial[blockIdx.x] = s;
    }
}

__global__ __launch_bounds__(1024)
void pose_loss_stage2(const float* __restrict__ partial, float* __restrict__ out, int n)
{
    __shared__ double sw[32];
    const int t = threadIdx.x;
    double v = (t < n) ? (double)partial[t] : 0.0;
#pragma unroll
    for (int off = 16; off > 0; off >>= 1)
        v += __shfl_down(v, off, 32);
    if ((t & 31) == 0) sw[t >> 5] = v;
    __syncthreads();
    if (t == 0) {
        double s = 0.0;
        for (int i = 0; i < 32; ++i) s += sw[i];
        out[0] = (float)s;
    }
}

extern "C" void kernel_launch(void* const* d_in, const int* in_sizes, int n_in,
                              void* d_out, int out_size, void* d_ws, size_t ws_size,
                              hipStream_t stream) {
    (void)in_sizes; (void)n_in; (void)out_size; (void)ws_size;
    const float* pose   = (const float*)d_in[0];
    const float* bone2d = (const float*)d_in[1];
    const float* lift   = (const float*)d_in[2];
    const float* Rd     = (const float*)d_in[3];
    const float* Cd     = (const float*)d_in[4];
    const float* blen   = (const float*)d_in[5];
    const float* Rcam   = (const float*)d_in[6];
    const int*   conn   = (const int*)d_in[7];
    float* out     = (float*)d_out;
    float* partial = (float*)d_ws;   // NBLK floats of scratch

    pose_loss_stage1<<<NBLK, BLK, 0, stream>>>(pose, bone2d, lift, Rd, Cd,
                                               blen, Rcam, conn, partial);
    pose_loss_stage2<<<1, 1024, 0, stream>>>(partial, out, NBLK);
}